// TAHIN_52458730553634
// MI455X (gfx1250) — compile-verified
//
#include <hip/hip_runtime.h>

#define N_USERS 100000
#define N_ITEMS 50000
#define N_NODES (N_USERS + N_ITEMS)
#define DIM 64
#define NI 128
#define N_LAYERS 2
#define E_ALL 3200000
#define E_U 1600000
#define E_I 800000
#define N_MP 2

typedef __attribute__((ext_vector_type(2))) float v2f;
typedef __attribute__((ext_vector_type(8))) float v8f;

// ---------------------------------------------------------------- elementwise
__global__ void zero_kernel(float* __restrict__ p, int n) {
  for (int i = blockIdx.x * blockDim.x + threadIdx.x; i < n;
       i += gridDim.x * blockDim.x)
    p[i] = 0.f;
}

__global__ void init_kernel(const float* __restrict__ fu,
                            const float* __restrict__ fi,
                            float* __restrict__ total,
                            float* __restrict__ cur) {
  const int n = N_NODES * DIM;
  for (int i = blockIdx.x * blockDim.x + threadIdx.x; i < n;
       i += gridDim.x * blockDim.x) {
    float v = (i < N_USERS * DIM) ? fu[i] : fi[i - N_USERS * DIM];
    total[i] = v;
    cur[i] = v;
  }
}

// ------------------------------------------------------- DCCF spmm scatter-add
// y[dst] += val * x[src], 16 threads per edge (float4 gather, 4 f32 atomics)
__global__ void spmm_scatter_kernel(const int* __restrict__ dst,
                                    const int* __restrict__ src,
                                    const float* __restrict__ vals,
                                    const float* __restrict__ x,
                                    float* __restrict__ y) {
  const int nt = E_ALL * 16;
  for (int t = blockIdx.x * blockDim.x + threadIdx.x; t < nt;
       t += gridDim.x * blockDim.x) {
    const int e = t >> 4, q = t & 15;
    const int d = dst[e], s = src[e];
    const float v = vals[e];
    const float4 xv = *(const float4*)(x + (size_t)s * DIM + q * 4);
    float* yp = y + (size_t)d * DIM + q * 4;
    atomicAdd(yp + 0, v * xv.x);
    atomicAdd(yp + 1, v * xv.y);
    atomicAdd(yp + 2, v * xv.z);
    atomicAdd(yp + 3, v * xv.w);
  }
}

// ------------------------------------------------- DCCF intent update (WMMA)
// For rows [row_offset, row_offset+n_rows):
//   P = softmax_row( X @ U );  Y = P @ U^T
//   cur = gnn + Y + cur_old;   total += cur
__global__ __launch_bounds__(128) void intent_update_kernel(
    float* __restrict__ cur, const float* __restrict__ gnn,
    float* __restrict__ total, const float* __restrict__ U, int row_offset,
    int n_rows) {
  __shared__ float sU[DIM * NI];       // 32 KB  U (64x128)
  __shared__ float sX[4][16 * DIM];    // 16 KB  per-wave X tile
  __shared__ float sL[4][16 * NI];     // 32 KB  per-wave logits / probs

  const int tid = threadIdx.x;
  const int wave = tid >> 5;
  const int lane = tid & 31;
  const int col = lane & 15;
  const int kh = lane >> 4;

  for (int i = tid; i < DIM * NI; i += 128) sU[i] = U[i];

  const int tile = blockIdx.x * 4 + wave;
  const int row0 = tile * 16;
  const bool active = row0 < n_rows;

  if (active) {
    for (int i = lane; i < 16 * DIM; i += 32) {
      const int r = i >> 6, c = i & 63;
      const int lr = row0 + r;
      sX[wave][i] = (lr < n_rows) ? cur[(size_t)(row_offset + lr) * DIM + c]
                                  : 0.f;
    }
  }
  __syncthreads();

  // GEMM1: (16x64) @ (64x128) via wmma_f32_16x16x4_f32
  if (active) {
    const float* xrow = &sX[wave][col * DIM];
    for (int nt = 0; nt < 8; ++nt) {
      v8f acc = {};
      for (int ks = 0; ks < 16; ++ks) {
        const int k0 = ks * 4 + 2 * kh;
        v2f a;
        a.x = xrow[k0];
        a.y = xrow[k0 + 1];
        v2f b;
        b.x = sU[k0 * NI + nt * 16 + col];
        b.y = sU[(k0 + 1) * NI + nt * 16 + col];
        acc = __builtin_amdgcn_wmma_f32_16x16x4_f32(false, a, false, b,
                                                    (short)0, acc, false,
                                                    false);
      }
#pragma unroll
      for (int v = 0; v < 8; ++v)
        sL[wave][(v + 8 * kh) * NI + nt * 16 + col] = acc[v];
    }
  }
  __syncthreads();

  // per-row softmax over the 128 intents
  if (active && lane < 16) {
    float* lrow = &sL[wave][lane * NI];
    float mx = lrow[0];
    for (int j = 1; j < NI; ++j) mx = fmaxf(mx, lrow[j]);
    float s = 0.f;
    for (int j = 0; j < NI; ++j) {
      const float ev = __expf(lrow[j] - mx);
      lrow[j] = ev;
      s += ev;
    }
    const float rs = 1.f / s;
    for (int j = 0; j < NI; ++j) lrow[j] *= rs;
  }
  __syncthreads();

  // GEMM2: (16x128) @ (128x64), B = U^T; fused residual writeback
  if (active) {
    const float* prow = &sL[wave][col * NI];
    for (int nt = 0; nt < 4; ++nt) {
      v8f acc = {};
      const int n = nt * 16 + col;  // output dim index 0..63
      for (int ks = 0; ks < 32; ++ks) {
        const int k0 = ks * 4 + 2 * kh;
        v2f a;
        a.x = prow[k0];
        a.y = prow[k0 + 1];
        v2f b;
        b.x = sU[n * NI + k0];      // U^T[k][n] = U[n][k]
        b.y = sU[n * NI + k0 + 1];
        acc = __builtin_amdgcn_wmma_f32_16x16x4_f32(false, a, false, b,
                                                    (short)0, acc, false,
                                                    false);
      }
#pragma unroll
      for (int v = 0; v < 8; ++v) {
        const int r = v + 8 * kh;
        if (row0 + r < n_rows) {
          const size_t idx = (size_t)(row_offset + row0 + r) * DIM + n;
          const float oldv = sX[wave][r * DIM + n];
          const float nc = gnn[idx] + acc[v] + oldv;
          cur[idx] = nc;
          total[idx] += nc;
        }
      }
    }
  }
}

// ---------------------------------------------------------------- HAN pieces
__global__ void deg_kernel(const int* __restrict__ src,
                           const int* __restrict__ dst,
                           float* __restrict__ deg_out,
                           float* __restrict__ deg_in, int ne) {
  for (int t = blockIdx.x * blockDim.x + threadIdx.x; t < ne;
       t += gridDim.x * blockDim.x) {
    atomicAdd(deg_out + src[t], 1.f);
    atomicAdd(deg_in + dst[t], 1.f);
  }
}

// z[dst] += rsqrt(deg_out[src]) * h[src]
__global__ void conv_scatter_kernel(const int* __restrict__ src,
                                    const int* __restrict__ dst,
                                    const float* __restrict__ deg_out,
                                    const float* __restrict__ h,
                                    float* __restrict__ z, int ne) {
  const int nt = ne * 16;
  for (int t = blockIdx.x * blockDim.x + threadIdx.x; t < nt;
       t += gridDim.x * blockDim.x) {
    const int e = t >> 4, q = t & 15;
    const int s = src[e], d = dst[e];
    const float inv = rsqrtf(fmaxf(deg_out[s], 1e-12f));
    const float4 xv = *(const float4*)(h + (size_t)s * DIM + q * 4);
    float* zp = z + (size_t)d * DIM + q * 4;
    atomicAdd(zp + 0, inv * xv.x);
    atomicAdd(zp + 1, inv * xv.y);
    atomicAdd(zp + 2, inv * xv.z);
    atomicAdd(zp + 3, inv * xv.w);
  }
}

__global__ void scale_kernel(float* __restrict__ z,
                             const float* __restrict__ deg_in, int n_rows) {
  const int n = n_rows * DIM;
  for (int i = blockIdx.x * blockDim.x + threadIdx.x; i < n;
       i += gridDim.x * blockDim.x) {
    const float dgv = deg_in[i >> 6];
    const float inv = dgv > 0.f ? rsqrtf(fmaxf(dgv, 1e-12f)) : 0.f;
    z[i] *= inv;
  }
}

// S_m += sum_n tanh(z[n] @ W1 + b1) . w2   (WMMA GEMM + fused reduction)
__global__ __launch_bounds__(128) void sem_score_kernel(
    const float* __restrict__ z, int n_rows, const float* __restrict__ W1,
    const float* __restrict__ b1, const float* __restrict__ w2,
    float* __restrict__ Sm) {
  __shared__ float sW[DIM * NI];     // 32 KB
  __shared__ float sZ[4][16 * DIM];  // 16 KB

  const int tid = threadIdx.x;
  const int wave = tid >> 5;
  const int lane = tid & 31;
  const int col = lane & 15;
  const int kh = lane >> 4;

  for (int i = tid; i < DIM * NI; i += 128) sW[i] = W1[i];

  const int tile = blockIdx.x * 4 + wave;
  const int row0 = tile * 16;
  const bool active = row0 < n_rows;

  if (active) {
    for (int i = lane; i < 16 * DIM; i += 32) {
      const int r = i >> 6, c = i & 63;
      const int lr = row0 + r;
      sZ[wave][i] = (lr < n_rows) ? z[(size_t)lr * DIM + c] : 0.f;
    }
  }
  __syncthreads();

  float partial = 0.f;
  if (active) {
    const float* xrow = &sZ[wave][col * DIM];
    for (int nt = 0; nt < 8; ++nt) {
      v8f acc = {};
      for (int ks = 0; ks < 16; ++ks) {
        const int k0 = ks * 4 + 2 * kh;
        v2f a;
        a.x = xrow[k0];
        a.y = xrow[k0 + 1];
        v2f b;
        b.x = sW[k0 * NI + nt * 16 + col];
        b.y = sW[(k0 + 1) * NI + nt * 16 + col];
        acc = __builtin_amdgcn_wmma_f32_16x16x4_f32(false, a, false, b,
                                                    (short)0, acc, false,
                                                    false);
      }
      const int c = nt * 16 + col;
      const float bb = b1[c], ww = w2[c];
#pragma unroll
      for (int v = 0; v < 8; ++v) {
        if (row0 + v + 8 * kh < n_rows) partial += tanhf(acc[v] + bb) * ww;
      }
    }
  }
  for (int off = 16; off > 0; off >>= 1)
    partial += __shfl_down(partial, off, 32);
  if (active && lane == 0) atomicAdd(Sm, partial);
}

__global__ void beta_kernel(const float* __restrict__ S, float inv_n,
                            float* __restrict__ beta) {
  if (threadIdx.x == 0 && blockIdx.x == 0) {
    const float w0 = S[0] * inv_n, w1 = S[1] * inv_n;
    const float m = fmaxf(w0, w1);
    const float e0 = __expf(w0 - m), e1 = __expf(w1 - m);
    const float d = e0 + e1;
    beta[0] = e0 / d;
    beta[1] = e1 / d;
  }
}

// out = 0.5*out_total + 0.5*(beta0*z0 + beta1*z1)
__global__ void combine_kernel(float* __restrict__ out,
                               const float* __restrict__ z0,
                               const float* __restrict__ z1,
                               const float* __restrict__ beta, int n) {
  const float b0 = beta[0], b1v = beta[1];
  for (int i = blockIdx.x * blockDim.x + threadIdx.x; i < n;
       i += gridDim.x * blockDim.x)
    out[i] = 0.5f * out[i] + 0.5f * (b0 * z0[i] + b1v * z1[i]);
}

// ------------------------------------------------------------------- launch
static inline unsigned gblk(long long n, int tpb) {
  return (unsigned)((n + tpb - 1) / tpb);
}

extern "C" void kernel_launch(void* const* d_in, const int* in_sizes, int n_in,
                              void* d_out, int out_size, void* d_ws,
                              size_t ws_size, hipStream_t stream) {
  (void)in_sizes; (void)n_in; (void)out_size; (void)ws_size;
  const int* Gidx = (const int*)d_in[0];
  const float* Gval = (const float*)d_in[1];
  const float* fu = (const float*)d_in[2];
  const float* fi = (const float*)d_in[3];
  const float* Uint = (const float*)d_in[4];
  const float* Iint = (const float*)d_in[5];
  const int* mpu = (const int*)d_in[6];
  const int* mpi = (const int*)d_in[7];
  const float* uW1 = (const float*)d_in[8];
  const float* ub1 = (const float*)d_in[9];
  const float* uw2 = (const float*)d_in[10];
  const float* iW1 = (const float*)d_in[11];
  const float* ib1 = (const float*)d_in[12];
  const float* iw2 = (const float*)d_in[13];
  float* out = (float*)d_out;  // also serves as `total`

  float* cur = (float*)d_ws;                         // N_NODES*64
  float* gnn = cur + (size_t)N_NODES * DIM;          // N_NODES*64
  float* deg = gnn + (size_t)N_NODES * DIM;          // 2*N_USERS
  float* S = deg + 2 * N_USERS;                      // 2
  float* beta = S + 2;                               // 2

  const int TPB = 256;
  const int nNode = N_NODES * DIM;

  init_kernel<<<gblk(nNode, TPB), TPB, 0, stream>>>(fu, fi, out, cur);

  // ---- DCCF: 2 layers
  const int uTiles = gblk(N_USERS / 16, 4);  // 1563 blocks of 4 waves
  const int iTiles = gblk(N_ITEMS / 16, 4);  // 782
  for (int l = 0; l < N_LAYERS; ++l) {
    zero_kernel<<<gblk(nNode, TPB), TPB, 0, stream>>>(gnn, nNode);
    spmm_scatter_kernel<<<gblk((long long)E_ALL * 16, TPB), TPB, 0, stream>>>(
        Gidx, Gidx + E_ALL, Gval, cur, gnn);
    intent_update_kernel<<<uTiles, 128, 0, stream>>>(cur, gnn, out, Uint, 0,
                                                     N_USERS);
    intent_update_kernel<<<iTiles, 128, 0, stream>>>(cur, gnn, out, Iint,
                                                     N_USERS, N_ITEMS);
  }

  // ---- HAN users (reuse cur[..] region as the two z planes)
  {
    float* z0 = cur;
    float* z1 = cur + (size_t)N_USERS * DIM;
    for (int m = 0; m < N_MP; ++m) {
      const int* src = mpu + (size_t)m * 2 * E_U;
      const int* dst = src + E_U;
      float* zm = (m == 0) ? z0 : z1;
      zero_kernel<<<gblk(2 * N_USERS, TPB), TPB, 0, stream>>>(deg,
                                                              2 * N_USERS);
      deg_kernel<<<gblk(E_U, TPB), TPB, 0, stream>>>(src, dst, deg,
                                                     deg + N_USERS, E_U);
      zero_kernel<<<gblk(N_USERS * DIM, TPB), TPB, 0, stream>>>(
          zm, N_USERS * DIM);
      conv_scatter_kernel<<<gblk((long long)E_U * 16, TPB), TPB, 0, stream>>>(
          src, dst, deg, fu, zm, E_U);
      scale_kernel<<<gblk(N_USERS * DIM, TPB), TPB, 0, stream>>>(
          zm, deg + N_USERS, N_USERS);
    }
    zero_kernel<<<1, 64, 0, stream>>>(S, 2);
    sem_score_kernel<<<uTiles, 128, 0, stream>>>(z0, N_USERS, uW1, ub1, uw2,
                                                 S + 0);
    sem_score_kernel<<<uTiles, 128, 0, stream>>>(z1, N_USERS, uW1, ub1, uw2,
                                                 S + 1);
    beta_kernel<<<1, 32, 0, stream>>>(S, 1.f / (float)N_USERS, beta);
    combine_kernel<<<gblk(N_USERS * DIM, TPB), TPB, 0, stream>>>(
        out, z0, z1, beta, N_USERS * DIM);
  }

  // ---- HAN items
  {
    float* z0 = cur;
    float* z1 = cur + (size_t)N_ITEMS * DIM;
    for (int m = 0; m < N_MP; ++m) {
      const int* src = mpi + (size_t)m * 2 * E_I;
      const int* dst = src + E_I;
      float* zm = (m == 0) ? z0 : z1;
      zero_kernel<<<gblk(2 * N_ITEMS, TPB), TPB, 0, stream>>>(deg,
                                                              2 * N_ITEMS);
      deg_kernel<<<gblk(E_I, TPB), TPB, 0, stream>>>(src, dst, deg,
                                                     deg + N_ITEMS, E_I);
      zero_kernel<<<gblk(N_ITEMS * DIM, TPB), TPB, 0, stream>>>(
          zm, N_ITEMS * DIM);
      conv_scatter_kernel<<<gblk((long long)E_I * 16, TPB), TPB, 0, stream>>>(
          src, dst, deg, fi, zm, E_I);
      scale_kernel<<<gblk(N_ITEMS * DIM, TPB), TPB, 0, stream>>>(
          zm, deg + N_ITEMS, N_ITEMS);
    }
    zero_kernel<<<1, 64, 0, stream>>>(S, 2);
    sem_score_kernel<<<iTiles, 128, 0, stream>>>(z0, N_ITEMS, iW1, ib1, iw2,
                                                 S + 0);
    sem_score_kernel<<<iTiles, 128, 0, stream>>>(z1, N_ITEMS, iW1, ib1, iw2,
                                                 S + 1);
    beta_kernel<<<1, 32, 0, stream>>>(S, 1.f / (float)N_ITEMS, beta);
    combine_kernel<<<gblk(N_ITEMS * DIM, TPB), TPB, 0, stream>>>(
        out + (size_t)N_USERS * DIM, z0, z1, beta, N_ITEMS * DIM);
  }
}